// GIN_7327214207515
// MI455X (gfx1250) — compile-verified
//
#include <hip/hip_runtime.h>
#include <hip/hip_bf16.h>

#define N_NODES 50000
#define N_EDGES 640000
#define DIM     128
#define NLAYERS 3
#define BN_EPS  1e-5f

typedef __attribute__((ext_vector_type(2))) float v2f;
typedef __attribute__((ext_vector_type(8))) float v8f;
typedef __attribute__((ext_vector_type(4))) int   v4i;

// ---------- CDNA5 async global->LDS copy (ASYNCcnt-tracked), with fallback ----
#if __has_builtin(__builtin_amdgcn_global_load_async_to_lds_b128)
#define HAVE_ASYNC_LDS 1
typedef __attribute__((address_space(1))) v4i* g_v4i_ptr;   // global
typedef __attribute__((address_space(3))) v4i* l_v4i_ptr;   // LDS
#else
#define HAVE_ASYNC_LDS 0
#endif

__device__ __forceinline__ void async_g2l_b128(const float* __restrict__ g,
                                               float* __restrict__ l) {
#if HAVE_ASYNC_LDS
    __builtin_amdgcn_global_load_async_to_lds_b128(
        (g_v4i_ptr)(void*)g,          // src: global, v4i*
        (l_v4i_ptr)(void*)l,          // dst: LDS,    v4i*
        /*offset=*/0, /*cpol=*/0);
#else
    *(float4*)l = *(const float4*)g;  // synchronous fallback
#endif
}

__device__ __forceinline__ void wait_async_lds() {
#if HAVE_ASYNC_LDS
#if __has_builtin(__builtin_amdgcn_s_wait_asynccnt)
    __builtin_amdgcn_s_wait_asynccnt(0);
#else
    asm volatile("s_wait_asynccnt 0" ::: "memory");
#endif
#endif
}

// ---------------- utility: zero a float buffer ----------------
__global__ void gin_zero_f32(float* __restrict__ p, int n) {
    int i = blockIdx.x * blockDim.x + threadIdx.x;
    if (i < n) p[i] = 0.0f;
}

// ---------------- edge scatter: agg[dst] += h[src] ----------------
__global__ __launch_bounds__(256)
void gin_scatter(const float* __restrict__ h,
                 const int* __restrict__ src,
                 const int* __restrict__ dst,
                 float* __restrict__ agg) {
    int idx = blockIdx.x * 256 + threadIdx.x;     // < E*128 = 81.92M < 2^31
    int e = idx >> 7;
    int d = idx & 127;
    if (e < N_EDGES) {
        int s = src[e];
        int t = dst[e];
        atomicAdd(&agg[t * DIM + d], h[s * DIM + d]);
    }
}

// ---------------- GEMM1: z = (h+agg) @ W1 + b1, fused BN-stat accumulation ----
// 256 threads = 8 waves; block owns a 16-row strip, wave w owns columns
// [16w,16w+16). Raw h/agg tiles staged to LDS via async copies; the "+" is
// fused into the LDS->VGPR A load. K loop: V_WMMA_F32_16X16X4_F32.
__global__ __launch_bounds__(256)
void gin_gemm1_bnstat(const float* __restrict__ h,
                      const float* __restrict__ agg,
                      const float* __restrict__ W,     // [K=128][N=128]
                      const float* __restrict__ bias,  // [128]
                      float* __restrict__ z,
                      float* __restrict__ bnsum,
                      float* __restrict__ bnsq) {
    __shared__ float sH[16 * 132];                 // padded stride 132
    __shared__ float sG[16 * 132];
    const int tid = threadIdx.x;
    const int rowBase = blockIdx.x * 16;

    // 16 rows x 128 cols = 512 x b128 chunks per buffer; 2 chunks/thread/buffer
    #pragma unroll
    for (int i = 0; i < 2; ++i) {
        int chunk = tid + i * 256;                 // 0..511
        int r  = chunk >> 5;                       // 32 chunks per row
        int c4 = (chunk & 31) * 4;
        int g  = (rowBase + r) * DIM + c4;
        int l  = r * 132 + c4;
        async_g2l_b128(h   + g, sH + l);
        async_g2l_b128(agg + g, sG + l);
    }
    wait_async_lds();
    __syncthreads();

    const int wave = tid >> 5;                     // 0..7 -> 16-col tile
    const int lane = tid & 31;
    const int half = lane >> 4;                    // 0 | 1
    const int ln16 = lane & 15;
    const int col  = wave * 16 + ln16;

    v8f acc = {0.f,0.f,0.f,0.f,0.f,0.f,0.f,0.f};
    #pragma unroll
    for (int k = 0; k < DIM; k += 4) {
        int ka = k + 2 * half;                     // lanes 0-15: K=k,k+1 ; 16-31: K=k+2,k+3
        int ab = ln16 * 132 + ka;
        v2f a, b;
        a.x = sH[ab]     + sG[ab];
        a.y = sH[ab + 1] + sG[ab + 1];
        b.x = W[ka * DIM + col];
        b.y = W[(ka + 1) * DIM + col];
        acc = __builtin_amdgcn_wmma_f32_16x16x4_f32(false, a, false, b,
                                                    (short)0, acc, false, false);
    }

    const float bv = bias[col];
    float s = 0.f, q = 0.f;
    #pragma unroll
    for (int v = 0; v < 8; ++v) {
        float o = acc[v] + bv;
        int row = rowBase + 8 * half + v;          // C layout: M = v + 8*half
        z[row * DIM + col] = o;
        s += o;
        q += o * o;
    }
    // lanes l and l+16 hold the same column -> combine, one atomic per column
    s += __shfl_xor(s, 16, 32);
    q += __shfl_xor(q, 16, 32);
    if (half == 0) {
        atomicAdd(&bnsum[col], s);
        atomicAdd(&bnsq[col],  q);
    }
}

// ---------------- BN finalize: scale/shift per column ----------------
__global__ void gin_bnfinal(const float* __restrict__ bnsum,
                            const float* __restrict__ bnsq,
                            const float* __restrict__ gamma,
                            const float* __restrict__ beta,
                            float* __restrict__ scale,
                            float* __restrict__ shift) {
    int c = threadIdx.x;
    if (c < DIM) {
        const float invN = 1.0f / (float)N_NODES;
        float mu  = bnsum[c] * invN;
        float var = bnsq[c] * invN - mu * mu;      // biased variance (torch BN)
        float sc  = gamma[c] * rsqrtf(var + BN_EPS);
        scale[c] = sc;
        shift[c] = beta[c] - mu * sc;
    }
}

// ---------------- GEMM2: out = [relu]( relu(z*scale+shift) @ W2 + b2 ) --------
// Raw z tile staged async; BN apply + ReLU fused into the LDS->VGPR A load
// using a small LDS scale/shift table (broadcast reads).
__global__ __launch_bounds__(256)
void gin_gemm2(const float* __restrict__ z,
               const float* __restrict__ scale,
               const float* __restrict__ shift,
               const float* __restrict__ W,
               const float* __restrict__ bias,
               float* __restrict__ out,
               int relu_out) {
    __shared__ float sZ[16 * 132];
    __shared__ float sScale[DIM];
    __shared__ float sShift[DIM];
    const int tid = threadIdx.x;
    const int rowBase = blockIdx.x * 16;

    #pragma unroll
    for (int i = 0; i < 2; ++i) {
        int chunk = tid + i * 256;
        int r  = chunk >> 5;
        int c4 = (chunk & 31) * 4;
        async_g2l_b128(z + (rowBase + r) * DIM + c4, sZ + r * 132 + c4);
    }
    if (tid < DIM)            sScale[tid]       = scale[tid];
    else if (tid < 2 * DIM)   sShift[tid - DIM] = shift[tid - DIM];
    wait_async_lds();
    __syncthreads();

    const int wave = tid >> 5;
    const int lane = tid & 31;
    const int half = lane >> 4;
    const int ln16 = lane & 15;
    const int col  = wave * 16 + ln16;

    v8f acc = {0.f,0.f,0.f,0.f,0.f,0.f,0.f,0.f};
    #pragma unroll
    for (int k = 0; k < DIM; k += 4) {
        int ka = k + 2 * half;
        int ab = ln16 * 132 + ka;
        v2f a, b;
        a.x = fmaxf(fmaf(sZ[ab],     sScale[ka],     sShift[ka]),     0.0f);
        a.y = fmaxf(fmaf(sZ[ab + 1], sScale[ka + 1], sShift[ka + 1]), 0.0f);
        b.x = W[ka * DIM + col];
        b.y = W[(ka + 1) * DIM + col];
        acc = __builtin_amdgcn_wmma_f32_16x16x4_f32(false, a, false, b,
                                                    (short)0, acc, false, false);
    }

    const float bv = bias[col];
    #pragma unroll
    for (int v = 0; v < 8; ++v) {
        float o = acc[v] + bv;
        if (relu_out) o = fmaxf(o, 0.0f);
        int row = rowBase + 8 * half + v;
        out[row * DIM + col] = o;
    }
}

// ---------------- host-side orchestration ----------------
extern "C" void kernel_launch(void* const* d_in, const int* in_sizes, int n_in,
                              void* d_out, int out_size, void* d_ws, size_t ws_size,
                              hipStream_t stream) {
    const float* x     = (const float*)d_in[0];
    const float* W1    = (const float*)d_in[1];   // [L][128][128] (in,out)
    const float* b1    = (const float*)d_in[2];   // [L][128]
    const float* gamma = (const float*)d_in[3];
    const float* beta  = (const float*)d_in[4];
    const float* W2    = (const float*)d_in[5];
    const float* b2    = (const float*)d_in[6];
    const int*   ei    = (const int*)d_in[7];     // [2][E]
    const int* src = ei;
    const int* dst = ei + N_EDGES;

    const size_t ND = (size_t)N_NODES * DIM;      // 6.4M floats
    float* agg   = (float*)d_ws;
    float* zbuf  = agg  + ND;
    float* hbuf  = zbuf + ND;
    float* bnsum = hbuf + ND;                      // 128
    float* bnsq  = bnsum + DIM;                    // 128
    float* scale = bnsq  + DIM;                    // 128
    float* shift = scale + DIM;                    // 128

    const dim3 blk(256);
    const int nd_i = (int)ND;

    for (int i = 0; i < NLAYERS; ++i) {
        const float* hin  = (i == 0) ? x : hbuf;
        float*       hout = (i == NLAYERS - 1) ? (float*)d_out : hbuf;

        gin_zero_f32<<<(nd_i + 255) / 256, blk, 0, stream>>>(agg, nd_i);
        gin_zero_f32<<<1, blk, 0, stream>>>(bnsum, 2 * DIM);

        gin_scatter<<<(N_EDGES * DIM) / 256, blk, 0, stream>>>(hin, src, dst, agg);

        gin_gemm1_bnstat<<<N_NODES / 16, blk, 0, stream>>>(
            hin, agg, W1 + (size_t)i * DIM * DIM, b1 + i * DIM, zbuf, bnsum, bnsq);

        gin_bnfinal<<<1, dim3(DIM), 0, stream>>>(bnsum, bnsq,
                                                 gamma + i * DIM, beta + i * DIM,
                                                 scale, shift);

        gin_gemm2<<<N_NODES / 16, blk, 0, stream>>>(
            zbuf, scale, shift, W2 + (size_t)i * DIM * DIM, b2 + i * DIM,
            hout, (i < NLAYERS - 1) ? 1 : 0);
    }
}